// YOLOLoss_8675833938056
// MI455X (gfx1250) — compile-verified
//
#include <hip/hip_runtime.h>

// YOLO loss for MI455X (gfx1250, wave32). Pure HBM-bound streaming reduction:
//   reads 88.3 MB of fp32 pred once (~3.8us floor @ 23.3 TB/s), emits 1 float.
// WMMA (V_WMMA_F32_16X16X4_F32, all-ones A) performs the exact fp32 32-lane
// wave reduction; block partials -> d_ws -> deterministic single-block fold.

#define A_ 3
#define C_ 80
#define G_ 52
#define B_ 32
#define T_ 32
#define G2 (G_ * G_)                       // 2704
#define CH (5 + C_)                        // 85
#define BLOCK 256
#define CHUNKS ((G2 + BLOCK - 1) / BLOCK)  // 11
#define NBLOCKS (B_ * A_ * CHUNKS)         // 1056

typedef __attribute__((ext_vector_type(2))) float v2f;
typedef __attribute__((ext_vector_type(8))) float v8f;

// Exact fp32 sum of all 32 lane values, via WMMA f32 16x16x4 with A = ones.
// B(4x16) holds one lane value per slot (b.y = 0 fills the other 32 slots),
// so D[m][n] = sum_k B[k][n] for every m. c[0] across the wave holds the 16
// column sums (rows M=0 and M=8 are identical), so xor-folding offsets
// 1,2,4,8 gives the full 32-lane total in every lane. EXEC must be all ones:
// call sites are unconditional for the whole block.
__device__ __forceinline__ float wave_sum_wmma(float s) {
  v2f a; a.x = 1.0f; a.y = 1.0f;   // A = ones(16x4)
  v2f b; b.x = s;    b.y = 0.0f;   // one distinct B slot per lane
  v8f c = {};
  c = __builtin_amdgcn_wmma_f32_16x16x4_f32(false, a, false, b, (short)0, c,
                                            false, false);
  float v = c[0];
  v += __shfl_xor(v, 1, 32);
  v += __shfl_xor(v, 2, 32);
  v += __shfl_xor(v, 4, 32);
  v += __shfl_xor(v, 8, 32);
  return v;  // every lane holds the wave total
}

__global__ __launch_bounds__(BLOCK) void yolo_main_kernel(
    const float* __restrict__ pred,     // (B, A*85, G, G)
    const float* __restrict__ boxes,    // (B, T, 4)
    const int*   __restrict__ classes,  // (B, T)
    float* __restrict__ partial) {      // (NBLOCKS,)
  __shared__ float sbx[T_ * 4];
  __shared__ int   scl[T_];
  __shared__ float swave[BLOCK / 32];

  const int blk   = blockIdx.x;
  const int slice = blk / CHUNKS;   // b*A + a
  const int chunk = blk % CHUNKS;
  const int b     = slice / A_;
  const int tid   = threadIdx.x;

  if (tid < T_) {
    const float* bp = boxes + (size_t)(b * T_ + tid) * 4;
    sbx[tid * 4 + 0] = bp[0];
    sbx[tid * 4 + 1] = bp[1];
    sbx[tid * 4 + 2] = bp[2];
    sbx[tid * 4 + 3] = bp[3];
    scl[tid] = classes[b * T_ + tid];
  }
  __syncthreads();

  const int q = chunk * BLOCK + tid;   // pixel index AND cls-row index
  float s = 0.0f;
  if (q < G2) {
    const int gx = q % G_;
    const int gy = q / G_;

    // Rebuild the scattered targets for this cell (collision-free scatter;
    // "last write wins" preserved by not breaking out of the loop).
    float tx = 0.f, ty = 0.f, tw = 0.f, th = 0.f, tobj = 0.f;
    int lbl = 0;
    #pragma unroll
    for (int t = 0; t < T_; ++t) {
      float bx = sbx[t * 4 + 0] * (float)G_;
      float by = sbx[t * 4 + 1] * (float)G_;
      int gi = (int)floorf(bx);
      int gj = (int)floorf(by);
      if (gi == gx && gj == gy) {
        tx = bx - (float)gi; ty = by - (float)gj;
        tw = sbx[t * 4 + 2]; th = sbx[t * 4 + 3];
        tobj = 1.0f; lbl = scl[t];
      }
    }

    const float* base = pred + (size_t)slice * (CH * G2);

    // bbox: channels 0..3, lane-coalesced loads (stride 1 in q across wave)
    float d0 = base[0 * G2 + q] - tx;
    float d1 = base[1 * G2 + q] - ty;
    float d2 = base[2 * G2 + q] - tw;
    float d3 = base[3 * G2 + q] - th;
    s += d0 * d0 + d1 * d1 + d2 * d2 + d3 * d3;

    // obj: BCE with logits, channel 4
    float x = base[4 * G2 + q];
    s += fmaxf(x, 0.f) - x * tobj + log1pf(__expf(-fabsf(x)));

    // cls: reshape(-1, C) rows are 80 CONTIGUOUS floats of the cls region
    // (channel boundaries straddled on purpose — matches the reference's
    // reshape of the (C,G,G) flat layout). 16B-aligned float4 streaming.
    const float* row = base + 5 * G2 + (size_t)q * C_;
    float4 v[C_ / 4];
    #pragma unroll
    for (int k = 0; k < C_ / 4; ++k) v[k] = ((const float4*)row)[k];
    float m = -__builtin_inff();
    #pragma unroll
    for (int k = 0; k < C_ / 4; ++k)
      m = fmaxf(m, fmaxf(fmaxf(v[k].x, v[k].y), fmaxf(v[k].z, v[k].w)));
    float se = 0.f;
    #pragma unroll
    for (int k = 0; k < C_ / 4; ++k)
      se += __expf(v[k].x - m) + __expf(v[k].y - m) +
            __expf(v[k].z - m) + __expf(v[k].w - m);
    float xl = row[lbl];                  // label picks row element j = label
    s += m + __logf(se) - xl;             // logsumexp(row) - row[label]
  }

  // Block reduction: WMMA wave sum (EXEC full here), then LDS fold.
  float w = wave_sum_wmma(s);
  if ((tid & 31) == 0) swave[tid >> 5] = w;
  __syncthreads();
  if (tid == 0) {
    float tot = 0.f;
    #pragma unroll
    for (int i = 0; i < BLOCK / 32; ++i) tot += swave[i];
    partial[blk] = tot;
  }
}

__global__ __launch_bounds__(BLOCK) void yolo_finalize_kernel(
    const float* __restrict__ partial, float* __restrict__ out) {
  __shared__ float swave[BLOCK / 32];
  float s = 0.f;
  for (int i = threadIdx.x; i < NBLOCKS; i += BLOCK) s += partial[i];
  float w = wave_sum_wmma(s);
  if ((threadIdx.x & 31) == 0) swave[threadIdx.x >> 5] = w;
  __syncthreads();
  if (threadIdx.x == 0) {
    float tot = 0.f;
    #pragma unroll
    for (int i = 0; i < BLOCK / 32; ++i) tot += swave[i];
    out[0] = tot / (float)B_;
  }
}

extern "C" void kernel_launch(void* const* d_in, const int* in_sizes, int n_in,
                              void* d_out, int out_size, void* d_ws,
                              size_t ws_size, hipStream_t stream) {
  const float* pred    = (const float*)d_in[0];
  const float* boxes   = (const float*)d_in[1];
  const int*   classes = (const int*)d_in[2];
  float* partial = (float*)d_ws;   // NBLOCKS floats (~4.2 KB), rewritten fully
  yolo_main_kernel<<<NBLOCKS, BLOCK, 0, stream>>>(pred, boxes, classes,
                                                  partial);
  yolo_finalize_kernel<<<1, BLOCK, 0, stream>>>(partial, (float*)d_out);
}